// MOIPool_77094662963361
// MI455X (gfx1250) — compile-verified
//
#include <hip/hip_runtime.h>
#include <hip/hip_bf16.h>

// ---------------- problem constants (from reference) ----------------
constexpr int kN   = 4;
constexpr int kC   = 256;
constexpr int kH   = 56;
constexpr int kW   = 56;
constexpr int kHW  = kH * kW;        // 3136
constexpr int kB   = 128;
constexpr int kS   = 200;
constexpr int kPH  = 7;
constexpr int kPW  = 7;
constexpr int kNB  = kPH * kPW;      // 49
constexpr int kCPB = 32;             // channels per block
constexpr float kSCALE = 0.0625f;

#define AS1 __attribute__((address_space(1)))
#define AS3 __attribute__((address_space(3)))

typedef __attribute__((ext_vector_type(4))) int v4i_t;

// ---------------- async-to-LDS helpers (CDNA5 path) ----------------
__device__ __forceinline__ void async_b128_to_lds(const void* gptr, void* lptr) {
#if __has_builtin(__builtin_amdgcn_global_load_async_to_lds_b128)
    __builtin_amdgcn_global_load_async_to_lds_b128(
        (AS1 v4i_t*)(gptr), (AS3 v4i_t*)(lptr), 0, 0);
#else
    unsigned long long ga = (unsigned long long)gptr;
    unsigned int la = (unsigned int)(unsigned long long)(AS3 void*)lptr;
    asm volatile("global_load_async_to_lds_b128 %0, %1, off"
                 :: "v"(la), "v"(ga) : "memory");
#endif
}

__device__ __forceinline__ void wait_asynccnt0() {
#if __has_builtin(__builtin_amdgcn_s_wait_asynccnt)
    __builtin_amdgcn_s_wait_asynccnt(0);
#else
    asm volatile("s_wait_asynccnt 0" ::: "memory");
#endif
}

// ---------------- kernel 1: per-(roi, pixel) superpixel mask ----------------
// mask[b*kHW + p] = oh_labels[b][ superpixels[bidx[b]][p] ] > 0
__global__ __launch_bounds__(256) void moi_mask_kernel(
    const float* __restrict__ rois,       // [B,5]
    const float* __restrict__ oh,         // [B,S]
    const int*   __restrict__ sp,         // [N,H,W]
    unsigned char* __restrict__ mask)     // [B, H*W]
{
    int i = blockIdx.x * blockDim.x + threadIdx.x;
    if (i >= kB * kHW) return;
    int b = i / kHW;
    int p = i - b * kHW;
    int bi = (int)rois[b * 5 + 0];
    int s  = sp[bi * kHW + p];
    mask[i] = (oh[b * kS + s] > 0.0f) ? (unsigned char)1 : (unsigned char)0;
}

// ---------------- kernel 2: masked ROI max-pool + argmax ----------------
// grid = (B, C/kCPB), block = 256 (8 wave32s). Mask plane (3136B) is staged
// into LDS once per block with async b128 transfers and reused by 32 channels.
__global__ __launch_bounds__(256) void moi_pool_kernel(
    const float* __restrict__ inp,        // [N,C,H,W]
    const float* __restrict__ rois,       // [B,5]
    const unsigned char* __restrict__ mask, // [B,H*W]
    float* __restrict__ out,              // [B,C,7,7]
    float* __restrict__ argf)             // [B,C,7,7] (argmax as float)
{
    __shared__ __align__(16) unsigned char smask[kHW];

    const int b  = blockIdx.x;
    const int c0 = blockIdx.y * kCPB;
    const int t  = threadIdx.x;

    // ---- stage per-ROI mask plane into LDS via async tensor path ----
    {
        const uint4* g = (const uint4*)(mask + (size_t)b * kHW);
        uint4*       l = (uint4*)smask;
        if (t < kHW / 16) {                 // 196 x 16B transfers
            async_b128_to_lds((const void*)(g + t), (void*)(l + t));
        }
        wait_asynccnt0();
    }
    __syncthreads();

    // ---- ROI geometry (uniform per block; jnp.round == rint, half-to-even) ----
    const int bi = (int)rois[b * 5 + 0];
    const int x1 = (int)rintf(rois[b * 5 + 1] * kSCALE);
    const int y1 = (int)rintf(rois[b * 5 + 2] * kSCALE);
    const int x2 = (int)rintf(rois[b * 5 + 3] * kSCALE);
    const int y2 = (int)rintf(rois[b * 5 + 4] * kSCALE);
    int rw = x2 - x1 + 1; if (rw < 1) rw = 1;
    int rh = y2 - y1 + 1; if (rh < 1) rh = 1;
    const float bw = (float)rw / (float)kPW;
    const float bh = (float)rh / (float)kPH;

    const float* fbatch = inp + (size_t)bi * kC * kHW;

    for (int idx = t; idx < kCPB * kNB; idx += 256) {
        const int c   = c0 + idx / kNB;
        const int bin = idx - (idx / kNB) * kNB;
        const int ph  = bin / kPW;
        const int pw  = bin - ph * kPW;

        int hs = (int)floorf((float)ph * bh) + y1;
        int he = (int)ceilf ((float)(ph + 1) * bh) + y1;
        int ws = (int)floorf((float)pw * bw) + x1;
        int we = (int)ceilf ((float)(pw + 1) * bw) + x1;
        hs = hs < 0 ? 0 : (hs > kH ? kH : hs);
        he = he < 0 ? 0 : (he > kH ? kH : he);
        ws = ws < 0 ? 0 : (ws > kW ? kW : ws);
        we = we < 0 ? 0 : (we > kW ? kW : we);

        const float* f = fbatch + (size_t)c * kHW;
        float best = -INFINITY;
        int   barg = -1;

        for (int h = hs; h < he; ++h) {
            const unsigned char* mrow = smask + h * kW;   // ds_load_u8 from LDS
            const float*         frow = f + h * kW;
            if (h + 1 < he) {
                __builtin_prefetch(frow + kW + ws, 0, 1); // global_prefetch_b8
            }
            for (int w = ws; w < we; ++w) {
                if (mrow[w]) {
                    float v = frow[w];
                    if (v > best) { best = v; barg = h * kW + w; }
                }
            }
        }

        const int o = (b * kC + c) * kNB + bin;
        out[o]  = (barg >= 0) ? best : 0.0f;
        argf[o] = (float)barg;
    }
}

extern "C" void kernel_launch(void* const* d_in, const int* in_sizes, int n_in,
                              void* d_out, int out_size, void* d_ws, size_t ws_size,
                              hipStream_t stream) {
    const float* inp  = (const float*)d_in[0];   // (4,256,56,56) f32
    const float* rois = (const float*)d_in[1];   // (128,5)       f32
    const float* oh   = (const float*)d_in[2];   // (128,200)     f32
    const int*   sp   = (const int*)d_in[3];     // (4,56,56)     int

    unsigned char* mask = (unsigned char*)d_ws;  // [B, H*W] = 401,408 B

    float* out  = (float*)d_out;                       // B*C*49 pooled values
    float* argf = out + (size_t)kB * kC * kNB;         // B*C*49 argmax (as float)

    {
        int total = kB * kHW;
        moi_mask_kernel<<<(total + 255) / 256, 256, 0, stream>>>(rois, oh, sp, mask);
    }
    {
        dim3 grid(kB, kC / kCPB);   // (128, 8) blocks
        moi_pool_kernel<<<grid, 256, 0, stream>>>(inp, rois, mask, out, argf);
    }
}